// Proposed_model_54030688584251
// MI455X (gfx1250) — compile-verified
//
#include <hip/hip_runtime.h>

typedef __attribute__((ext_vector_type(2))) float v2f;
typedef __attribute__((ext_vector_type(8))) float v8f;

#define DIM 64
#define KC 10

// gfx1250 async global->LDS path (guarded: falls back to float4 staging if the
// toolchain lacks the builtins; host pass always takes the fallback).
#if defined(__AMDGCN__) && __has_builtin(__builtin_amdgcn_global_load_async_to_lds_b128)
#define HAVE_ASYNC_LDS 1
#define GAS __attribute__((address_space(1)))
#define LAS __attribute__((address_space(3)))
typedef int vb128 __attribute__((vector_size(4 * sizeof(int))));
__device__ __forceinline__ void wait_asynccnt0() {
#if __has_builtin(__builtin_amdgcn_s_wait_asynccnt)
  __builtin_amdgcn_s_wait_asynccnt(0);
#else
  asm volatile("s_wait_asynccnt 0" ::: "memory");
#endif
}
#endif

// ---------------------------------------------------------------- utilities

__global__ void k_fill(float* p, float v, size_t n) {
  size_t i = (size_t)blockIdx.x * blockDim.x + threadIdx.x;
  if (i < n) p[i] = v;
}

__device__ __forceinline__ void atomicMaxFloat(float* addr, float val) {
  if (val >= 0.0f) atomicMax((int*)addr, __float_as_int(val));
  else             atomicMin((unsigned int*)addr, (unsigned int)__float_as_int(val));
}

// ---------------------------------------------------------------- degrees

__global__ void k_deg(const int* __restrict__ src, const int* __restrict__ dst,
                      float* degs, float* degd, int E) {
  int i = blockIdx.x * blockDim.x + threadIdx.x;
  if (i < E) {
    atomicAdd(&degs[src[i]], 1.0f);
    atomicAdd(&degd[dst[i]], 1.0f);
  }
}

// ---------------------------------------------------------------- WMMA GEMM
// Y[N,64] = X@W (+ X2@W2) (+ bias) then optional row-scale deg^-1/2.
// One 16-row stripe per 128-thread block; wave w owns columns 16w..16w+15.
// f32 WMMA 16x16x4, K-loop of 16 steps; W (and A stripe) staged in LDS via
// the async-to-LDS DMA path when available.

__global__ __launch_bounds__(128)
void k_gemm64(const float* __restrict__ X, const float* __restrict__ W,
              const float* __restrict__ X2, const float* __restrict__ W2,
              const float* __restrict__ bias, const float* __restrict__ rowdeg,
              float* __restrict__ Y, int N) {
  __shared__ float sW[64 * 64];
  __shared__ float sW2[64 * 64];
  __shared__ float sA[16 * 64];
  __shared__ float sA2[16 * 64];
  const int tid = threadIdx.x;
  const int row0 = blockIdx.x * 16;
  const bool fullA = (row0 + 16 <= N);

#if HAVE_ASYNC_LDS
  {
    GAS vb128* gW = (GAS vb128*)W;
    LAS vb128* lW = (LAS vb128*)sW;
    for (int i = tid; i < 1024; i += 128)
      __builtin_amdgcn_global_load_async_to_lds_b128(gW + i, lW + i, 0, 0);
    if (X2) {
      GAS vb128* gW2 = (GAS vb128*)W2;
      LAS vb128* lW2 = (LAS vb128*)sW2;
      for (int i = tid; i < 1024; i += 128)
        __builtin_amdgcn_global_load_async_to_lds_b128(gW2 + i, lW2 + i, 0, 0);
    }
    if (fullA) {
      GAS vb128* gX = (GAS vb128*)(X + (size_t)row0 * 64);
      LAS vb128* lA = (LAS vb128*)sA;
      for (int i = tid; i < 256; i += 128)
        __builtin_amdgcn_global_load_async_to_lds_b128(gX + i, lA + i, 0, 0);
      if (X2) {
        GAS vb128* gX2 = (GAS vb128*)(X2 + (size_t)row0 * 64);
        LAS vb128* lA2 = (LAS vb128*)sA2;
        for (int i = tid; i < 256; i += 128)
          __builtin_amdgcn_global_load_async_to_lds_b128(gX2 + i, lA2 + i, 0, 0);
      }
    } else {
      float4* sAv = (float4*)sA;
      float4* sA2v = (float4*)sA2;
      for (int i = tid; i < 256; i += 128) {
        size_t base = (size_t)row0 * 64 + (size_t)i * 4;
        float4 v = {0.f, 0.f, 0.f, 0.f};
        if (base + 3 < (size_t)N * 64) v = *(const float4*)(X + base);
        sAv[i] = v;
        if (X2) {
          float4 v2 = {0.f, 0.f, 0.f, 0.f};
          if (base + 3 < (size_t)N * 64) v2 = *(const float4*)(X2 + base);
          sA2v[i] = v2;
        }
      }
    }
    wait_asynccnt0();
  }
#else
  {
    const float4* Wv = (const float4*)W;
    float4* sWv = (float4*)sW;
    for (int i = tid; i < 1024; i += 128) sWv[i] = Wv[i];
    if (X2) {
      const float4* W2v = (const float4*)W2;
      float4* sW2v = (float4*)sW2;
      for (int i = tid; i < 1024; i += 128) sW2v[i] = W2v[i];
    }
    float4* sAv = (float4*)sA;
    for (int i = tid; i < 256; i += 128) {
      size_t base = (size_t)row0 * 64 + (size_t)i * 4;
      float4 v = {0.f, 0.f, 0.f, 0.f};
      if (base + 3 < (size_t)N * 64) v = *(const float4*)(X + base);
      sAv[i] = v;
    }
    if (X2) {
      float4* sA2v = (float4*)sA2;
      for (int i = tid; i < 256; i += 128) {
        size_t base = (size_t)row0 * 64 + (size_t)i * 4;
        float4 v = {0.f, 0.f, 0.f, 0.f};
        if (base + 3 < (size_t)N * 64) v = *(const float4*)(X2 + base);
        sA2v[i] = v;
      }
    }
  }
#endif
  __syncthreads();

  const int lane = tid & 31;
  const int wid  = tid >> 5;
  const int lrow = lane & 15;   // M (for A) / N (for B)
  const int lhi  = lane >> 4;   // K sub-pair select
  const int colb = wid * 16;

  v8f c = {};
#pragma unroll
  for (int k = 0; k < 64; k += 4) {
    const int kk = k + lhi * 2;
    v2f a, b;
    a.x = sA[lrow * 64 + kk];
    a.y = sA[lrow * 64 + kk + 1];
    b.x = sW[kk * 64 + colb + lrow];
    b.y = sW[(kk + 1) * 64 + colb + lrow];
    c = __builtin_amdgcn_wmma_f32_16x16x4_f32(false, a, false, b, (short)0, c, false, false);
  }
  if (X2) {
#pragma unroll
    for (int k = 0; k < 64; k += 4) {
      const int kk = k + lhi * 2;
      v2f a, b;
      a.x = sA2[lrow * 64 + kk];
      a.y = sA2[lrow * 64 + kk + 1];
      b.x = sW2[kk * 64 + colb + lrow];
      b.y = sW2[(kk + 1) * 64 + colb + lrow];
      c = __builtin_amdgcn_wmma_f32_16x16x4_f32(false, a, false, b, (short)0, c, false, false);
    }
  }

  const int col = colb + lrow;
  const float bv = bias ? bias[col] : 0.0f;
#pragma unroll
  for (int i = 0; i < 8; i++) {
    int r = row0 + lhi * 8 + i;
    if (r < N) {
      float v = c[i] + bv;
      if (rowdeg) { float dv = rowdeg[r]; v *= (dv > 0.0f ? rsqrtf(dv) : 0.0f); }
      Y[(size_t)r * 64 + col] = v;
    }
  }
}

// ---------------------------------------------------------------- edge scatter
// wave per edge: gather row M[src]*w, atomic-add into out[dst], count dst.

__global__ void k_scatter(const float* __restrict__ M, const int* __restrict__ src,
                          const int* __restrict__ dst, const float* __restrict__ ew,
                          float* __restrict__ out, float* __restrict__ cnt, int E) {
  int e = blockIdx.x * 8 + (threadIdx.x >> 5);
  int lane = threadIdx.x & 31;
  if (e >= E) return;
  int s = src[e], d = dst[e];
  float w = ew ? ew[e] : 1.0f;
  const float2 v = *(const float2*)(M + (size_t)s * 64 + lane * 2);
  atomicAdd(out + (size_t)d * 64 + lane * 2 + 0, v.x * w);
  atomicAdd(out + (size_t)d * 64 + lane * 2 + 1, v.y * w);
  if (cnt && lane == 0) atomicAdd(cnt + d, 1.0f);
}

__global__ void k_scale_deg(float* Y, const float* __restrict__ deg, size_t n64) {
  size_t i = (size_t)blockIdx.x * blockDim.x + threadIdx.x;
  if (i < n64) { float dv = deg[i >> 6]; Y[i] *= (dv > 0.0f ? rsqrtf(dv) : 0.0f); }
}

__global__ void k_div_cnt(float* Y, const float* __restrict__ cnt, size_t n64) {
  size_t i = (size_t)blockIdx.x * blockDim.x + threadIdx.x;
  if (i < n64) Y[i] /= fmaxf(cnt[i >> 6], 1.0f);
}

// ---------------------------------------------------------------- GAT attention

__global__ void k_rowdot2(const float* __restrict__ F, const float* __restrict__ al,
                          const float* __restrict__ ar, float* el, float* er, int N) {
  int r = blockIdx.x * 8 + (threadIdx.x >> 5);
  int lane = threadIdx.x & 31;
  if (r >= N) return;
  float2 f = *(const float2*)(F + (size_t)r * 64 + lane * 2);
  float2 a = *(const float2*)(al + lane * 2);
  float2 b = *(const float2*)(ar + lane * 2);
  float s1 = f.x * a.x + f.y * a.y;
  float s2 = f.x * b.x + f.y * b.y;
  for (int o = 16; o > 0; o >>= 1) { s1 += __shfl_down(s1, o, 32); s2 += __shfl_down(s2, o, 32); }
  if (lane == 0) { el[r] = s1; er[r] = s2; }
}

__global__ void k_attn_e(const float* __restrict__ el, const float* __restrict__ er,
                         const int* __restrict__ src, const int* __restrict__ dst,
                         float* evals, float* m, int E) {
  int e = blockIdx.x * blockDim.x + threadIdx.x;
  if (e < E) {
    float v = el[src[e]] + er[dst[e]];
    v = v > 0.0f ? v : 0.2f * v;            // leaky_relu slope 0.2
    evals[e] = v;
    atomicMaxFloat(m + dst[e], v);
  }
}

__global__ void k_attn_exp(const float* __restrict__ m, const int* __restrict__ dst,
                           float* evals, float* s, int E) {
  int e = blockIdx.x * blockDim.x + threadIdx.x;
  if (e < E) {
    float ex = expf(evals[e] - m[dst[e]]);
    evals[e] = ex;
    atomicAdd(s + dst[e], ex);
  }
}

__global__ void k_attn_norm(const float* __restrict__ s, const int* __restrict__ dst,
                            float* evals, int E) {
  int e = blockIdx.x * blockDim.x + threadIdx.x;
  if (e < E) evals[e] /= s[dst[e]];
}

// ---------------------------------------------------------------- k-means

__global__ void k_km_assign(const float* __restrict__ X, const float* __restrict__ C,
                            int* __restrict__ labels, int G) {
  __shared__ float sC[KC * DIM];
  for (int i = threadIdx.x; i < KC * DIM; i += blockDim.x) sC[i] = C[i];
  __syncthreads();
  int g = blockIdx.x * blockDim.x + threadIdx.x;
  if (g >= G) return;
  float best = 3.4e38f; int bk = 0;
  for (int k = 0; k < KC; k++) {
    float a = 0.0f;
    for (int d = 0; d < DIM; d++) {
      float diff = X[(size_t)g * DIM + d] - sC[k * DIM + d];
      a += diff * diff;
    }
    if (a < best) { best = a; bk = k; }
  }
  labels[g] = bk;
}

__global__ void k_km_accum(const float* __restrict__ X, const int* __restrict__ labels,
                           float* __restrict__ csum, float* __restrict__ ccnt, int G) {
  __shared__ float ls[KC * DIM];
  __shared__ float lc[KC];
  for (int i = threadIdx.x; i < KC * DIM; i += blockDim.x) ls[i] = 0.0f;
  if (threadIdx.x < KC) lc[threadIdx.x] = 0.0f;
  __syncthreads();
  int g = blockIdx.x * blockDim.x + threadIdx.x;
  if (g < G) {
    int l = labels[g];
    atomicAdd(&lc[l], 1.0f);
    for (int d = 0; d < DIM; d++) atomicAdd(&ls[l * DIM + d], X[(size_t)g * DIM + d]);
  }
  __syncthreads();
  for (int i = threadIdx.x; i < KC * DIM; i += blockDim.x) atomicAdd(&csum[i], ls[i]);
  if (threadIdx.x < KC) atomicAdd(&ccnt[threadIdx.x], lc[threadIdx.x]);
}

__global__ void k_km_update(float* __restrict__ C, const float* __restrict__ csum,
                            const float* __restrict__ ccnt, int n) {
  int i = blockIdx.x * blockDim.x + threadIdx.x;
  if (i < n) C[i] = csum[i] / fmaxf(ccnt[i >> 6], 1.0f);
}

// ------------------------------------------------ diversity + final combine
// user_embed = 0.7*(u + 0.5*hud) + 0.2*agg + 0.1*soc
// (the later "- hud*keep*mask" term is identically zero since keep = !mask)

__global__ void k_div_combine(const float* __restrict__ UE, const float* __restrict__ C,
                              const float* __restrict__ agg, const float* __restrict__ soc,
                              float* __restrict__ outUE, int N) {
  __shared__ float sC[KC * DIM];
  __shared__ float sc2[KC];
  for (int i = threadIdx.x; i < KC * DIM; i += blockDim.x) sC[i] = C[i];
  __syncthreads();
  if (threadIdx.x < KC) {
    float a = 0.0f;
    for (int d = 0; d < DIM; d++) { float c = sC[threadIdx.x * DIM + d]; a += c * c; }
    sc2[threadIdx.x] = a;
  }
  __syncthreads();
  int u = blockIdx.x * blockDim.x + threadIdx.x;
  if (u >= N) return;
  float dot[KC];
#pragma unroll
  for (int k = 0; k < KC; k++) dot[k] = 0.0f;
  float du2 = 0.0f;
  for (int d = 0; d < DIM; d++) {
    float x = UE[(size_t)u * DIM + d];
    du2 += x * x;
#pragma unroll
    for (int k = 0; k < KC; k++) dot[k] += x * sC[k * DIM + d];
  }
  float w[KC], mx = -3.4e38f;
#pragma unroll
  for (int k = 0; k < KC; k++) {
    float v = du2 - 2.0f * dot[k] + sc2[k];
    w[k] = sqrtf(fmaxf(v, 1e-12f));
    mx = fmaxf(mx, w[k]);
  }
  float se = 0.0f;
#pragma unroll
  for (int k = 0; k < KC; k++) { w[k] = expf(w[k] - mx); se += w[k]; }
  float inv = 1.0f / se;
  for (int d = 0; d < DIM; d++) {
    float hud = 0.0f;
#pragma unroll
    for (int k = 0; k < KC; k++) hud += w[k] * inv * sC[k * DIM + d];
    size_t idx = (size_t)u * DIM + d;
    outUE[idx] = 0.7f * (UE[idx] + 0.5f * hud) + 0.2f * agg[idx] + 0.1f * soc[idx];
  }
}

// ---------------------------------------------------------------- losses

__global__ void k_gtcd(const float* __restrict__ IE, const float* __restrict__ C,
                       const int* __restrict__ labels, float* out, int G) {
  int g = blockIdx.x * blockDim.x + threadIdx.x;
  if (g >= G) return;
  int l = labels[g];
  float a = 0.0f;
  for (int d = 0; d < DIM; d++) {
    float diff = IE[(size_t)g * DIM + d] - C[l * DIM + d];
    a += diff * diff;
  }
  out[g] = sqrtf(a);
}

__global__ void k_play_loss(const float* __restrict__ UE, const float* __restrict__ C,
                            const int* __restrict__ labels, const int* __restrict__ pu,
                            const int* __restrict__ pg, float* total, float* cntp, int E) {
  int e = blockIdx.x * 8 + (threadIdx.x >> 5);
  int lane = threadIdx.x & 31;
  if (e >= E) return;
  int u = pu[e];
  int l = labels[pg[e]];
  float2 a = *(const float2*)(UE + (size_t)u * DIM + lane * 2);
  float2 c = *(const float2*)(C + (size_t)l * DIM + lane * 2);
  float dx = a.x - c.x, dy = a.y - c.y;
  float s = dx * dx + dy * dy;
  for (int o = 16; o > 0; o >>= 1) s += __shfl_down(s, o, 32);
  if (lane == 0) { atomicAdd(total + u, s); atomicAdd(cntp + u, 1.0f); }
}

__global__ void k_divloss(const float* __restrict__ total, const float* __restrict__ cntp,
                          float* dl, int U) {
  int u = blockIdx.x * blockDim.x + threadIdx.x;
  if (u < U) dl[u] = total[u] / (cntp[u] + 1e-10f);
}

// ------------------------------- k-th order-statistic via 32-step bisection
// monotone float->uint key; find key of sorted_desc[K]

__device__ __forceinline__ unsigned fkey(float f) {
  unsigned b = __float_as_uint(f);
  return (b & 0x80000000u) ? ~b : (b | 0x80000000u);
}

__global__ void k_sel_init(unsigned* st) {
  st[0] = 0u; st[1] = 0xFFFFFFFFu; st[2] = 0x7FFFFFFFu; st[3] = 0u;
}

__global__ void k_sel_count(const float* __restrict__ dl, unsigned* st, int U) {
  int i = blockIdx.x * blockDim.x + threadIdx.x;
  if (i < U && fkey(dl[i]) > st[2]) atomicAdd(&st[3], 1u);
}

__global__ void k_sel_step(unsigned* st, unsigned K) {
  if (st[3] <= K) st[1] = st[2]; else st[0] = st[2] + 1u;
  st[2] = st[0] + ((st[1] - st[0]) >> 1);
  st[3] = 0u;
}

__global__ void k_sel_fin(const unsigned* st, float* thr) {
  unsigned k = st[0];
  unsigned b = (k & 0x80000000u) ? (k ^ 0x80000000u) : ~k;
  thr[0] = __uint_as_float(b);
}

__global__ void k_mask(const float* __restrict__ dl, const float* __restrict__ thr,
                       float* maskOut, float* acc, int U) {
  int u = blockIdx.x * blockDim.x + threadIdx.x;
  if (u >= U) return;
  bool m = dl[u] > thr[0];
  maskOut[u] = m ? 1.0f : 0.0f;
  if (!m) { atomicAdd(acc + 0, dl[u]); atomicAdd(acc + 1, 1.0f); }
}

__global__ void k_fdl(const float* __restrict__ acc, float* out) {
  out[0] = acc[0] / fmaxf(acc[1], 1.0f);
}

// ---------------------------------------------------------------- driver

extern "C" void kernel_launch(void* const* d_in, const int* in_sizes, int n_in,
                              void* d_out, int out_size, void* d_ws, size_t ws_size,
                              hipStream_t stream) {
  const float* h_game0  = (const float*)d_in[0];
  const float* user_emb = (const float*)d_in[1];
  const float* item_emb = (const float*)d_in[2];
  const float* play_w   = (const float*)d_in[3];
  const float* W_gc0    = (const float*)d_in[4];
  const float* W_gc1    = (const float*)d_in[5];
  const float* Wi_self  = (const float*)d_in[6];
  const float* Wi_neigh = (const float*)d_in[7];
  const float* bi       = (const float*)d_in[8];
  const float* W_gat    = (const float*)d_in[9];
  const float* attn_l   = (const float*)d_in[10];
  const float* attn_r   = (const float*)d_in[11];
  const float* Ws_self  = (const float*)d_in[12];
  const float* Ws_neigh = (const float*)d_in[13];
  const float* bs       = (const float*)d_in[14];
  const int* item_src   = (const int*)d_in[15];
  const int* item_dst   = (const int*)d_in[16];
  const int* play_user  = (const int*)d_in[17];
  const int* play_game  = (const int*)d_in[18];
  const int* soc_src    = (const int*)d_in[19];
  const int* soc_dst    = (const int*)d_in[20];

  const int G = in_sizes[0] / DIM;
  const int U = in_sizes[1] / DIM;
  const int E_ITEM = in_sizes[15];
  const int E_PLAY = in_sizes[17];
  const int E_SOC  = in_sizes[19];
  const size_t UD = (size_t)U * DIM;
  const size_t GD = (size_t)G * DIM;

  // output layout: user_embed | item_embedding | fdl | gtcd | mask
  float* out = (float*)d_out;
  float* o_ue   = out;
  float* o_item = out + UD;
  float* o_fdl  = o_item + GD;
  float* o_gtcd = o_fdl + 1;
  float* o_mask = o_gtcd + G;

  // workspace carve-up
  float* ws = (float*)d_ws;
  size_t off = 0;
  auto alloc = [&](size_t n) { float* p = ws + off; off += n; return p; };
  float* buf1   = alloc(UD);            // hW / h_neigh_play / feat / h_user_social
  float* buf2   = alloc(UD);            // h_game / h_neigh_soc
  float* buf3   = alloc(UD);            // h_user_aggregate
  float* e_soc  = alloc((size_t)E_SOC); // e values -> exp -> alpha (in place)
  float* el     = alloc(U);
  float* er     = alloc(U);
  float* mseg   = alloc(U);
  float* sseg   = alloc(U);
  float* cnt    = alloc(U);
  float* total  = alloc(U);
  float* cntp   = alloc(U);
  float* dl     = alloc(U);
  float* degs   = alloc(G);
  float* degd   = alloc(G);
  int*   labels = (int*)alloc(G);
  float* centers = alloc(KC * DIM);
  float* csum    = alloc(KC * DIM);     // csum+ccnt contiguous (one fill)
  float* ccnt    = alloc(16);
  float* thr     = alloc(1);
  float* acc     = alloc(2);
  unsigned* selst = (unsigned*)alloc(4);

  auto gr = [](size_t n, int b) { return dim3((unsigned)((n + (size_t)b - 1) / (size_t)b)); };
  const dim3 B256(256);

  // ---- item graph degrees
  k_fill<<<gr(2 * (size_t)G, 256), B256, 0, stream>>>(degs, 0.0f, 2 * (size_t)G); // degs+degd contiguous
  k_deg<<<gr(E_ITEM, 256), B256, 0, stream>>>(item_src, item_dst, degs, degd, E_ITEM);

  // ---- GraphConv layer 1: buf1 = (h_game0 @ W_gc0) * ns ; scatter -> buf2 ; * nd
  k_gemm64<<<gr(G, 16), dim3(128), 0, stream>>>(h_game0, W_gc0, nullptr, nullptr, nullptr, degs, buf1, G);
  k_fill<<<gr(GD, 256), B256, 0, stream>>>(buf2, 0.0f, GD);
  k_scatter<<<gr((size_t)E_ITEM * 32, 256), B256, 0, stream>>>(buf1, item_src, item_dst, nullptr, buf2, nullptr, E_ITEM);
  k_scale_deg<<<gr(GD, 256), B256, 0, stream>>>(buf2, degd, GD);

  // ---- GraphConv layer 2
  k_gemm64<<<gr(G, 16), dim3(128), 0, stream>>>(buf2, W_gc1, nullptr, nullptr, nullptr, degs, buf1, G);
  k_fill<<<gr(GD, 256), B256, 0, stream>>>(buf2, 0.0f, GD);
  k_scatter<<<gr((size_t)E_ITEM * 32, 256), B256, 0, stream>>>(buf1, item_src, item_dst, nullptr, buf2, nullptr, E_ITEM);
  k_scale_deg<<<gr(GD, 256), B256, 0, stream>>>(buf2, degd, GD);     // buf2 = h_game

  // ---- play SAGE: buf1 = mean_w(h_game over play edges) ; buf3 = u@Wi_self + buf1@Wi_neigh + bi
  k_fill<<<gr(UD, 256), B256, 0, stream>>>(buf1, 0.0f, UD);
  k_fill<<<gr(U, 256), B256, 0, stream>>>(cnt, 0.0f, U);
  k_scatter<<<gr((size_t)E_PLAY * 32, 256), B256, 0, stream>>>(buf2, play_game, play_user, play_w, buf1, cnt, E_PLAY);
  k_div_cnt<<<gr(UD, 256), B256, 0, stream>>>(buf1, cnt, UD);
  k_gemm64<<<gr(U, 16), dim3(128), 0, stream>>>(user_emb, Wi_self, buf1, Wi_neigh, bi, nullptr, buf3, U);

  // ---- GAT attention on social graph
  k_gemm64<<<gr(U, 16), dim3(128), 0, stream>>>(buf3, W_gat, nullptr, nullptr, nullptr, nullptr, buf1, U); // feat
  k_rowdot2<<<gr((size_t)U * 32, 256), B256, 0, stream>>>(buf1, attn_l, attn_r, el, er, U);
  k_fill<<<gr(U, 256), B256, 0, stream>>>(mseg, -3.402823466e38f, U);
  k_fill<<<gr(U, 256), B256, 0, stream>>>(sseg, 0.0f, U);
  k_attn_e<<<gr(E_SOC, 256), B256, 0, stream>>>(el, er, soc_src, soc_dst, e_soc, mseg, E_SOC);
  k_attn_exp<<<gr(E_SOC, 256), B256, 0, stream>>>(mseg, soc_dst, e_soc, sseg, E_SOC);
  k_attn_norm<<<gr(E_SOC, 256), B256, 0, stream>>>(sseg, soc_dst, e_soc, E_SOC);  // e_soc = alpha

  // ---- social SAGE: buf2 = mean_alpha(user_emb) ; buf1 = u@Ws_self + buf2@Ws_neigh + bs
  k_fill<<<gr(UD, 256), B256, 0, stream>>>(buf2, 0.0f, UD);
  k_fill<<<gr(U, 256), B256, 0, stream>>>(cnt, 0.0f, U);
  k_scatter<<<gr((size_t)E_SOC * 32, 256), B256, 0, stream>>>(user_emb, soc_src, soc_dst, e_soc, buf2, cnt, E_SOC);
  k_div_cnt<<<gr(UD, 256), B256, 0, stream>>>(buf2, cnt, UD);
  k_gemm64<<<gr(U, 16), dim3(128), 0, stream>>>(user_emb, Ws_self, buf2, Ws_neigh, bs, nullptr, buf1, U); // h_user_social

  // ---- k-means on item_embedding (init = first 10 rows)
  (void)hipMemcpyAsync(centers, item_emb, KC * DIM * sizeof(float), hipMemcpyDeviceToDevice, stream);
  for (int it = 0; it < 10; ++it) {
    k_km_assign<<<gr(G, 256), B256, 0, stream>>>(item_emb, centers, labels, G);
    k_fill<<<gr(KC * DIM + 16, 256), B256, 0, stream>>>(csum, 0.0f, KC * DIM + 16);
    k_km_accum<<<gr(G, 256), B256, 0, stream>>>(item_emb, labels, csum, ccnt, G);
    k_km_update<<<gr(KC * DIM, 256), B256, 0, stream>>>(centers, csum, ccnt, KC * DIM);
  }
  k_km_assign<<<gr(G, 256), B256, 0, stream>>>(item_emb, centers, labels, G);

  // ---- diversity embedding + final combine -> o_ue
  k_div_combine<<<gr(U, 256), B256, 0, stream>>>(user_emb, centers, buf3, buf1, o_ue, U);

  // ---- pass-through item embedding
  (void)hipMemcpyAsync(o_item, item_emb, GD * sizeof(float), hipMemcpyDeviceToDevice, stream);

  // ---- game-to-cluster distances
  k_gtcd<<<gr(G, 256), B256, 0, stream>>>(item_emb, centers, labels, o_gtcd, G);

  // ---- per-user diversity losses over play edges
  k_fill<<<gr(2 * (size_t)U, 256), B256, 0, stream>>>(total, 0.0f, 2 * (size_t)U); // total+cntp contiguous
  k_play_loss<<<gr((size_t)E_PLAY * 32, 256), B256, 0, stream>>>(o_ue, centers, labels, play_user, play_game, total, cntp, E_PLAY);
  k_divloss<<<gr(U, 256), B256, 0, stream>>>(total, cntp, dl, U);

  // ---- threshold = sorted_desc[int(U*0.1)] via 32-step bisection on key space
  const unsigned Kth = (unsigned)((double)U * 0.1);
  k_sel_init<<<1, 1, 0, stream>>>(selst);
  for (int it = 0; it < 32; ++it) {
    k_sel_count<<<gr(U, 256), B256, 0, stream>>>(dl, selst, U);
    k_sel_step<<<1, 1, 0, stream>>>(selst, Kth);
  }
  k_sel_fin<<<1, 1, 0, stream>>>(selst, thr);

  // ---- mask, filtered loss (note: user_embed correction term is identically 0)
  k_fill<<<gr(2, 256), B256, 0, stream>>>(acc, 0.0f, 2);
  k_mask<<<gr(U, 256), B256, 0, stream>>>(dl, thr, o_mask, acc, U);
  k_fdl<<<1, 1, 0, stream>>>(acc, o_fdl);
}